// Camera_86346022518815
// MI455X (gfx1250) — compile-verified
//
#include <hip/hip_runtime.h>

#define HH 192
#define WW 256
#define SPP 8
#define NRAYS (HH * WW * SPP)   // 393216
#define NS 32
#define MAXD 6

typedef float v8f __attribute__((ext_vector_type(8)));
typedef float v2f __attribute__((ext_vector_type(2)));

// ---------------- JAX threefry2x32 (exact) ----------------
__device__ __forceinline__ void tf2x32(unsigned k0, unsigned k1,
                                       unsigned x0, unsigned x1,
                                       unsigned& o0, unsigned& o1) {
  unsigned ks2 = k0 ^ k1 ^ 0x1BD11BDAu;
  x0 += k0; x1 += k1;
#define TFR(r) { x0 += x1; x1 = (x1 << (r)) | (x1 >> (32 - (r))); x1 ^= x0; }
  TFR(13) TFR(15) TFR(26) TFR(6)   x0 += k1;  x1 += ks2 + 1u;
  TFR(17) TFR(29) TFR(16) TFR(24)  x0 += ks2; x1 += k0 + 2u;
  TFR(13) TFR(15) TFR(26) TFR(6)   x0 += k0;  x1 += k1 + 3u;
  TFR(17) TFR(29) TFR(16) TFR(24)  x0 += k1;  x1 += ks2 + 4u;
  TFR(13) TFR(15) TFR(26) TFR(6)   x0 += ks2; x1 += k0 + 5u;
#undef TFR
  o0 = x0; o1 = x1;
}

__device__ __forceinline__ float u01f(unsigned b) {
  return __uint_as_float((b >> 9) | 0x3f800000u) - 1.0f;
}

// XLA f32 erf_inv (Giles polynomial)
__device__ __forceinline__ float erfinv_f(float x) {
  float w = -logf((1.0f - x) * (1.0f + x));
  float p;
  if (w < 5.0f) {
    w -= 2.5f;
    p = 2.81022636e-08f;
    p = 3.43273939e-07f + p * w;
    p = -3.5233877e-06f + p * w;
    p = -4.39150654e-06f + p * w;
    p = 0.00021858087f + p * w;
    p = -0.00125372503f + p * w;
    p = -0.00417768164f + p * w;
    p = 0.246640727f + p * w;
    p = 1.50140941f + p * w;
  } else {
    w = sqrtf(w) - 3.0f;
    p = -0.000200214257f;
    p = 0.000100950558f + p * w;
    p = 0.00134934322f + p * w;
    p = -0.00367342844f + p * w;
    p = 0.00573950773f + p * w;
    p = -0.0076224613f + p * w;
    p = 0.00943887047f + p * w;
    p = 1.00167406f + p * w;
    p = 2.83297682f + p * w;
  }
  return p * x;
}

// jax.random.normal(key, (3,)): counts [0,1,2] padded -> pairs (0,2),(1,0)
__device__ __forceinline__ void normal3(unsigned k0, unsigned k1,
                                        float& n0, float& n1, float& n2) {
  unsigned a0, a1, b0, b1;
  tf2x32(k0, k1, 0u, 2u, a0, a1);
  tf2x32(k0, k1, 1u, 0u, b0, b1);
  const float lo = -0.99999994f;
  const float span = 1.0f - lo;  // rounds to 2.0f in f32, as in JAX
  float u;
  u = fmaxf(lo, u01f(a0) * span + lo); n0 = 1.41421356f * erfinv_f(u);
  u = fmaxf(lo, u01f(b0) * span + lo); n1 = 1.41421356f * erfinv_f(u);
  u = fmaxf(lo, u01f(a1) * span + lo); n2 = 1.41421356f * erfinv_f(u);
}

// _ball(key): kn,ku = split(key); v=normal(kn,(3,)); unit(v)*uniform(ku)^(1/3)
__device__ __forceinline__ void ball3(unsigned k0, unsigned k1,
                                      float& bx, float& by, float& bz) {
  unsigned a0, a1, b0, b1;
  tf2x32(k0, k1, 0u, 2u, a0, a1);
  tf2x32(k0, k1, 1u, 3u, b0, b1);
  float v0, v1, v2;
  normal3(a0, b0, v0, v1, v2);          // kn = (a0, b0)
  unsigned g0, g1;
  tf2x32(a1, b1, 0u, 0u, g0, g1);       // ku = (a1, b1), scalar uniform
  float u = fmaxf(0.0f, u01f(g0));
  float inv = rsqrtf(v0 * v0 + v1 * v1 + v2 * v2 + 1e-12f);
  float c = powf(u, 0.33333334f);
  bx = v0 * inv * c; by = v1 * inv * c; bz = v2 * inv * c;
}

// ---------------- Wave-cooperative 32-sphere hit via WMMA f32 16x16x4 ----------------
// A tile (per wave, constant): 16 spheres x K=4 (cx,cy,cz,0)
// B tile (built per call via shuffles): K=4 x 16 rays
// D tile: D[sphere][ray] = centers . vec
__device__ __forceinline__ void wave_hit(
    float ox, float oy, float oz, float dx, float dy, float dz, float tmin,
    const v2f (&A)[2], const float (&r2c)[2][8], const float (&cc2)[2][8],
    int lane, float& t_out, int& j_out) {
  const float INF = __builtin_inff();
  float od = ox * dx + oy * dy + oz * dz;
  float dd = dx * dx + dy * dy + dz * dz;
  float oo = ox * ox + oy * oy + oz * oz;
  bool hi = lane >= 16;
  float t_res = INF; int j_res = 0;
#pragma unroll
  for (int rT = 0; rT < 2; ++rT) {
    int src = (lane & 15) | (rT << 4);
    float sdx = __shfl(dx, src), sdy = __shfl(dy, src), sdz = __shfl(dz, src);
    float sox = __shfl(ox, src), soy = __shfl(oy, src), soz = __shfl(oz, src);
    float odc = __shfl(od, src), ddc = __shfl(dd, src), ooc = __shfl(oo, src);
    v2f Bd, Bo;
    Bd.x = hi ? sdz : sdx;  Bd.y = hi ? 0.0f : sdy;   // K = {0,1}|{2,3} per lane half
    Bo.x = hi ? soz : sox;  Bo.y = hi ? 0.0f : soy;
    float bt = INF; int bj = 0;
#pragma unroll
    for (int sT = 0; sT < 2; ++sT) {
      v8f zero = {};
      v8f cd = __builtin_amdgcn_wmma_f32_16x16x4_f32(false, A[sT], false, Bd,
                                                     (short)0, zero, false, false);
      v8f co = __builtin_amdgcn_wmma_f32_16x16x4_f32(false, A[sT], false, Bo,
                                                     (short)0, zero, false, false);
#pragma unroll
      for (int v = 0; v < 8; ++v) {
        float m = cd[v] - odc;                    // (c-o).d
        float tp = m / ddc;
        float d2 = (cc2[sT][v] - 2.0f * co[v] + ooc) - tp * m;  // |c-o|^2 - ((c-o).d)^2/dd
        float r2 = r2c[sT][v];
        bool inside = d2 < r2;
        float root = sqrtf(inside ? fmaxf(r2 - d2, 0.0f) : 1.0f);
        float tc = inside ? (tp - root) : INF;
        tc = ((tc > tmin) && __builtin_isfinite(tc)) ? tc : INF;
        int jc = sT * 16 + v + (hi ? 8 : 0);
        bool better = tc < bt;
        bj = better ? jc : bj;
        bt = better ? tc : bt;
      }
    }
    // combine partner lane (holds the other 8 sphere rows of this column)
    float pt = __shfl(bt, lane ^ 16);
    int   pj = __shfl(bj, lane ^ 16);
    bool take = (pt < bt) || ((pt == bt) && (pj < bj));  // argmin = first index
    bt = take ? pt : bt;  bj = take ? pj : bj;
    bool mine = ((lane >> 4) == rT);
    t_res = mine ? bt : t_res;  j_res = mine ? bj : j_res;
  }
  t_out = t_res;  j_out = j_res;
}

__device__ __forceinline__ void resolve_hit(
    float ox, float oy, float oz, float dx, float dy, float dz,
    float t, int j,
    const float* s_cx, const float* s_cy, const float* s_cz, const float* s_r,
    const float* s_al, const float* s_rf, const float* s_fz,
    const float* s_ri, const float* s_tr,
    float& px, float& py, float& pz,
    float& nx, float& ny, float& nz, bool& ff,
    float& al0, float& al1, float& al2,
    float& rf, float& fz, float& ri, float& tr) {
  bool hit = __builtin_isfinite(t);
  float ts = hit ? t : 0.0f;
  px = hit ? ox + ts * dx : 0.0f;
  py = hit ? oy + ts * dy : 0.0f;
  pz = hit ? oz + ts * dz : 0.0f;
  float cx = s_cx[j], cy = s_cy[j], cz = s_cz[j], rr = s_r[j];
  float owx = hit ? (px - cx) / rr : 0.0f;
  float owy = hit ? (py - cy) / rr : 0.0f;
  float owz = hit ? (pz - cz) / rr : 0.0f;
  ff = (dx * owx + dy * owy + dz * owz) < 0.0f;
  nx = ff ? owx : -owx;  ny = ff ? owy : -owy;  nz = ff ? owz : -owz;
  al0 = s_al[j * 3 + 0]; al1 = s_al[j * 3 + 1]; al2 = s_al[j * 3 + 2];
  rf = s_rf[j]; fz = s_fz[j]; ri = s_ri[j]; tr = s_tr[j];
}

__global__ __launch_bounds__(256) void pathtrace_kernel(
    const float* __restrict__ centers, const float* __restrict__ radii,
    const float* __restrict__ albedo,  const float* __restrict__ reflec,
    const float* __restrict__ fuzzi,   const float* __restrict__ rindex,
    const float* __restrict__ transp,  float* __restrict__ out) {
  __shared__ float s_cx[NS], s_cy[NS], s_cz[NS], s_r[NS], s_r2[NS], s_c2[NS];
  __shared__ float s_al[NS * 3], s_rf[NS], s_fz[NS], s_ri[NS], s_tr[NS];

  int tid = threadIdx.x;
  if (tid < NS) {
    float cx = centers[tid * 3 + 0], cy = centers[tid * 3 + 1], cz = centers[tid * 3 + 2];
    s_cx[tid] = cx; s_cy[tid] = cy; s_cz[tid] = cz;
    float rr = radii[tid];
    s_r[tid] = rr; s_r2[tid] = rr * rr;
    s_c2[tid] = cx * cx + cy * cy + cz * cz;
    s_al[tid * 3 + 0] = albedo[tid * 3 + 0];
    s_al[tid * 3 + 1] = albedo[tid * 3 + 1];
    s_al[tid * 3 + 2] = albedo[tid * 3 + 2];
    s_rf[tid] = reflec[tid]; s_fz[tid] = fuzzi[tid];
    s_ri[tid] = rindex[tid]; s_tr[tid] = transp[tid];
  }
  __syncthreads();

  int lane = tid & 31;
  int mrow = lane & 15;
  bool hi = lane >= 16;

  // Wave-constant WMMA A tiles + per-lane sphere constants (D-tile layout)
  v2f A[2];
  float r2c[2][8], cc2[2][8];
#pragma unroll
  for (int sT = 0; sT < 2; ++sT) {
    int j = sT * 16 + mrow;
    A[sT].x = hi ? s_cz[j] : s_cx[j];   // K=0 | K=2
    A[sT].y = hi ? 0.0f    : s_cy[j];   // K=1 | K=3 (pad)
#pragma unroll
    for (int v = 0; v < 8; ++v) {
      int jj = sT * 16 + v + (hi ? 8 : 0);
      r2c[sT][v] = s_r2[jj];
      cc2[sT][v] = s_c2[jj];
    }
  }

  // ----- per-ray key: row r of split(key(0), N) -----
  const unsigned N = (unsigned)NRAYS;
  unsigned r = (unsigned)(blockIdx.x * blockDim.x + tid);
  unsigned kr0, kr1;
  {
    unsigned i0 = 2u * r, i1 = 2u * r + 1u, a0, a1, b0, b1;
    if (r < (N >> 1)) {
      tf2x32(0u, 0u, i0, i0 + N, a0, a1);
      tf2x32(0u, 0u, i1, i1 + N, b0, b1);
      kr0 = a0; kr1 = b0;
    } else {
      tf2x32(0u, 0u, i0 - N, i0, a0, a1);
      tf2x32(0u, 0u, i1 - N, i1, b0, b1);
      kr0 = a1; kr1 = b1;
    }
  }
  // split(key_r) -> kray, kcol
  unsigned kray0, kray1, k0, k1;
  {
    unsigned a0, a1, b0, b1;
    tf2x32(kr0, kr1, 0u, 2u, a0, a1);
    tf2x32(kr0, kr1, 1u, 3u, b0, b1);
    kray0 = a0; kray1 = b0;  k0 = a1; k1 = b1;  // kcol
  }
  // pixel jitter: uniform(kray, (2,), -0.5, 0.5)
  float off0, off1;
  {
    unsigned g0, g1;
    tf2x32(kray0, kray1, 0u, 1u, g0, g1);
    off0 = fmaxf(-0.5f, u01f(g0) - 0.5f);
    off1 = fmaxf(-0.5f, u01f(g1) - 0.5f);
  }

  // ----- primary ray -----
  unsigned pix = r >> 3;
  float fi = (float)(pix % (unsigned)WW);
  float fj = (float)(pix / (unsigned)WW);
  const float sw  = 2.0f * ((float)WW / (float)HH);
  const float dux = sw / (float)WW;
  const float dvy = -2.0f / (float)HH;
  float psx = (-0.5f * sw + 0.5f * dux) + (fi + off0) * dux;
  float psy = (1.0f + 0.5f * dvy) + (fj + off1) * dvy;
  float psz = -1.0f;
  float len = sqrtf(psx * psx + psy * psy + psz * psz);
  float dx = psx / len, dy = psy / len, dz = psz / len;
  float ox = 0.0f, oy = 0.0f, oz = 0.0f;

  float att0 = 1.0f, att1 = 1.0f, att2 = 1.0f;
  int depth = MAXD;

  float t; int hj;
  wave_hit(ox, oy, oz, dx, dy, dz, 0.0f, A, r2c, cc2, lane, t, hj);
  float px, py, pz, nx, ny, nz; bool ff;
  float al0, al1, al2, rf, fz, ri, tr;
  resolve_hit(ox, oy, oz, dx, dy, dz, t, hj,
              s_cx, s_cy, s_cz, s_r, s_al, s_rf, s_fz, s_ri, s_tr,
              px, py, pz, nx, ny, nz, ff, al0, al1, al2, rf, fz, ri, tr);

  // ----- 6 uniform bounce steps (EXEC stays full for WMMA) -----
  for (int step = 0; step < MAXD; ++step) {
    bool active = (depth > 0) && (t > 0.0f) && __builtin_isfinite(t);
    // key_next, skey = split(key); k1,k2,k3 = split(skey, 3)
    unsigned a0, a1, b0, b1;
    tf2x32(k0, k1, 0u, 2u, a0, a1);
    tf2x32(k0, k1, 1u, 3u, b0, b1);
    unsigned kn0 = a0, kn1 = b0, sk0 = a1, sk1 = b1;
    unsigned f00, f01, f10, f11, f20, f21;
    tf2x32(sk0, sk1, 0u, 3u, f00, f01);
    tf2x32(sk0, sk1, 1u, 4u, f10, f11);
    tf2x32(sk0, sk1, 2u, 5u, f20, f21);
    // k1=(f00,f10)  k2=(f20,f01)  k3=(f11,f21)

    float b1x, b1y, b1z;
    ball3(f00, f10, b1x, b1y, b1z);
    float lbx = nx + b1x, lby = ny + b1y, lbz = nz + b1z;  // lambertian
    float dn = dx * nx + dy * ny + dz * nz;
    float rx = dx - 2.0f * dn * nx, ry = dy - 2.0f * dn * ny, rz = dz - 2.0f * dn * nz;
    float rinv = rsqrtf(rx * rx + ry * ry + rz * rz + 1e-12f);
    float b2x, b2y, b2z;
    ball3(f20, f01, b2x, b2y, b2z);
    float mx = rx * rinv + fz * b2x, my = ry * rinv + fz * b2y, mz = rz * rinv + fz * b2z;
    float drx = (1.0f - rf) * lbx + rf * mx;
    float dry = (1.0f - rf) * lby + rf * my;
    float drz = (1.0f - rf) * lbz + rf * mz;
    float eta = ff ? (1.0f / ri) : ri;
    float rpx = eta * (dx - dn * nx), rpy = eta * (dy - dn * ny), rpz = eta * (dz - dn * nz);
    float rp2 = rpx * rpx + rpy * rpy + rpz * rpz;
    float q = sqrtf(fmaxf(1.0f - rp2, 0.0f));
    float dtx = rpx - q * nx, dty = rpy - q * ny, dtz = rpz - q * nz;
    unsigned g0, g1;
    tf2x32(f11, f21, 0u, 0u, g0, g1);
    float u3 = fmaxf(0.0f, u01f(g0));
    bool takeR = u3 > tr;
    float cx_ = takeR ? drx : dtx, cy_ = takeR ? dry : dty, cz_ = takeR ? drz : dtz;
    float cinv = rsqrtf(cx_ * cx_ + cy_ * cy_ + cz_ * cz_ + 1e-12f);
    float ndx = cx_ * cinv, ndy = cy_ * cinv, ndz = cz_ * cinv;

    float nt; int nj;
    wave_hit(px, py, pz, ndx, ndy, ndz, 0.001f, A, r2c, cc2, lane, nt, nj);
    float npx, npy, npz, nnx, nny, nnz; bool nff;
    float nal0, nal1, nal2, nrf, nfz, nri, ntr;
    resolve_hit(px, py, pz, ndx, ndy, ndz, nt, nj,
                s_cx, s_cy, s_cz, s_r, s_al, s_rf, s_fz, s_ri, s_tr,
                npx, npy, npz, nnx, nny, nnz, nff,
                nal0, nal1, nal2, nrf, nfz, nri, ntr);
    int ndepth = depth - 1;
    float na0 = (ndepth > 0) ? al0 * att0 : 0.0f;
    float na1 = (ndepth > 0) ? al1 * att1 : 0.0f;
    float na2 = (ndepth > 0) ? al2 * att2 : 0.0f;
    if (active) {
      ox = px; oy = py; oz = pz;
      dx = ndx; dy = ndy; dz = ndz;
      att0 = na0; att1 = na1; att2 = na2;
      px = npx; py = npy; pz = npz;
      nx = nnx; ny = nny; nz = nnz;
      t = nt; ff = nff; hj = nj;
      al0 = nal0; al1 = nal1; al2 = nal2;
      rf = nrf; fz = nfz; ri = nri; tr = ntr;
      depth = ndepth;
    }
    k0 = kn0; k1 = kn1;
  }

  // sky + attenuation
  float dinv = rsqrtf(dx * dx + dy * dy + dz * dz + 1e-12f);
  float a = 0.5f * (dy * dinv + 1.0f);
  float c0 = att0 * ((1.0f - a) + a * 0.5f);
  float c1 = att1 * ((1.0f - a) + a * 0.7f);
  float c2 = att2 * ((1.0f - a) + a * 1.0f);

  // mean over 8 spp (lanes 8k..8k+7), gamma, clip, store
  c0 += __shfl_xor(c0, 1); c0 += __shfl_xor(c0, 2); c0 += __shfl_xor(c0, 4);
  c1 += __shfl_xor(c1, 1); c1 += __shfl_xor(c1, 2); c1 += __shfl_xor(c1, 4);
  c2 += __shfl_xor(c2, 1); c2 += __shfl_xor(c2, 2); c2 += __shfl_xor(c2, 4);
  if ((lane & 7) == 0) {
    float g;
    g = c0 * 0.125f; g = (g > 0.0f) ? sqrtf(g) : 0.0f; out[pix * 3 + 0] = fminf(g, 0.999f);
    g = c1 * 0.125f; g = (g > 0.0f) ? sqrtf(g) : 0.0f; out[pix * 3 + 1] = fminf(g, 0.999f);
    g = c2 * 0.125f; g = (g > 0.0f) ? sqrtf(g) : 0.0f; out[pix * 3 + 2] = fminf(g, 0.999f);
  }
}

extern "C" void kernel_launch(void* const* d_in, const int* in_sizes, int n_in,
                              void* d_out, int out_size, void* d_ws, size_t ws_size,
                              hipStream_t stream) {
  (void)in_sizes; (void)n_in; (void)out_size; (void)d_ws; (void)ws_size;
  pathtrace_kernel<<<NRAYS / 256, 256, 0, stream>>>(
      (const float*)d_in[0], (const float*)d_in[1], (const float*)d_in[2],
      (const float*)d_in[3], (const float*)d_in[4], (const float*)d_in[5],
      (const float*)d_in[6], (float*)d_out);
}